// MultiHeadAttention_5669356831537
// MI455X (gfx1250) — compile-verified
//
#include <hip/hip_runtime.h>
#include <hip/hip_bf16.h>

#define D_MODEL 1024
#define N_HEADS 16
#define D_KH    64
#define SEQ     2048
#define BATCH   4
#define NEGVAL  (-1000000000.0f)
#define LN_EPS  1e-5f

typedef __attribute__((ext_vector_type(2))) float v2f;
typedef __attribute__((ext_vector_type(8))) float v8f;
typedef int v4i_ __attribute__((vector_size(16)));  // matches builtin proto

#define AS_LD 40   // 32 + 8 pad (rows stay 16B aligned)
#define BS_LD 72   // 64 + 8 pad (rows stay 16B aligned)
#define ASZ   (64 * AS_LD)
#define BSZ   (32 * BS_LD)

// ---------------------------------------------------------------------------
// Async global->LDS copy (CDNA5 GLOBAL_LOAD_ASYNC_TO_LDS_B128, ASYNCcnt) with
// a synchronous float4 fallback if the builtin is unavailable.
// ---------------------------------------------------------------------------
#if defined(__HIP_DEVICE_COMPILE__) &&                                   \
    __has_builtin(__builtin_amdgcn_global_load_async_to_lds_b128) &&     \
    __has_builtin(__builtin_amdgcn_s_wait_asynccnt)
#define HAVE_ASYNC_LDS 1
#endif

__device__ __forceinline__ void cp_b128(float* lds_dst, const float* gsrc) {
#if defined(HAVE_ASYNC_LDS)
  __builtin_amdgcn_global_load_async_to_lds_b128(
      (__attribute__((address_space(1))) v4i_*)gsrc,
      (__attribute__((address_space(3))) v4i_*)lds_dst, 0, 0);
#else
  *(float4*)lds_dst = *(const float4*)gsrc;
#endif
}

template <int N>
__device__ __forceinline__ void cp_wait() {
#if defined(HAVE_ASYNC_LDS)
  __builtin_amdgcn_s_wait_asynccnt(N);
#endif
}

// ---------------------------------------------------------------------------
// Stage one K-step: A tile 64(m) x 32(k) from Asrc (leading dim lda), and
// B tile 32(k) x 64(n) from Bsrc (leading dim ldb). 8 B128 ops per thread.
// ---------------------------------------------------------------------------
__device__ __forceinline__ void stage_tiles(const float* __restrict__ Asrc,
                                            size_t lda,
                                            const float* __restrict__ Bsrc,
                                            size_t ldb, float* __restrict__ As,
                                            float* __restrict__ Bs, int tid) {
#pragma unroll
  for (int i = 0; i < 4; ++i) {
    const int idx = tid + i * 128;
    const int r = idx >> 3, c4 = idx & 7;
    cp_b128(As + r * AS_LD + c4 * 4, Asrc + (size_t)r * lda + c4 * 4);
  }
#pragma unroll
  for (int i = 0; i < 4; ++i) {
    const int idx = tid + i * 128;
    const int r = idx >> 4, c4 = idx & 15;
    cp_b128(Bs + r * BS_LD + c4 * 4, Bsrc + (size_t)r * ldb + c4 * 4);
  }
}

// ---------------------------------------------------------------------------
// Wave-level 32x32 tile via V_WMMA_F32_16X16X4_F32 (layouts per ISA 7.12.2).
// ---------------------------------------------------------------------------
__device__ __forceinline__ void wave_mma_32x32(const float* __restrict__ As,
                                               const float* __restrict__ Bs,
                                               v8f acc[2][2], int wr, int wc,
                                               int lr, int half) {
#pragma unroll
  for (int kk = 0; kk < 32; kk += 4) {
    v2f a[2], b[2];
#pragma unroll
    for (int mi = 0; mi < 2; ++mi) {
      const float* ap = As + (wr * 32 + mi * 16 + lr) * AS_LD + kk + 2 * half;
      a[mi].x = ap[0];
      a[mi].y = ap[1];
    }
#pragma unroll
    for (int ni = 0; ni < 2; ++ni) {
      const float* bp = Bs + (kk + 2 * half) * BS_LD + wc * 32 + ni * 16 + lr;
      b[ni].x = bp[0];
      b[ni].y = bp[BS_LD];
    }
#pragma unroll
    for (int mi = 0; mi < 2; ++mi)
#pragma unroll
      for (int ni = 0; ni < 2; ++ni)
        acc[mi][ni] = __builtin_amdgcn_wmma_f32_16x16x4_f32(
            false, a[mi], false, b[ni], (short)0, acc[mi][ni], false, false);
  }
}

// ---------------------------------------------------------------------------
// Double-buffered async mainloop shared by all GEMM kernels.
// Asrc points at (m0, k=0) with leading dim lda; Bsrc at (k=0, n0) with ldb.
// ---------------------------------------------------------------------------
__device__ __forceinline__ void gemm_mainloop(const float* __restrict__ Asrc,
                                              size_t lda,
                                              const float* __restrict__ Bsrc,
                                              size_t ldb, int nk,
                                              float* __restrict__ AsB,
                                              float* __restrict__ BsB,
                                              v8f acc[2][2], int tid, int wr,
                                              int wc, int lr, int half) {
  stage_tiles(Asrc, lda, Bsrc, ldb, AsB, BsB, tid);
  for (int kt = 0; kt < nk; ++kt) {
    const int cur = kt & 1;
    if (kt + 1 < nk) {
      stage_tiles(Asrc + (size_t)(kt + 1) * 32, lda,
                  Bsrc + (size_t)(kt + 1) * 32 * ldb, ldb,
                  AsB + (cur ^ 1) * ASZ, BsB + (cur ^ 1) * BSZ, tid);
      cp_wait<8>();  // previous stage's 8 ops have landed
    } else {
      cp_wait<0>();
    }
    __syncthreads();
    wave_mma_32x32(AsB + cur * ASZ, BsB + cur * BSZ, acc, wr, wc, lr, half);
    __syncthreads();  // protect buffer reuse by next issue
  }
}

// ---------------------------------------------------------------------------
// C[8192,1024] = (X @ W + bias)(+resid) * alpha
// mode 0: row-major out (+resid). mode 1: scatter [B,H,S,64]. mode 2: [B,H,64,S]
// ---------------------------------------------------------------------------
__global__ __launch_bounds__(128) void proj_gemm_kernel(
    const float* __restrict__ X, const float* __restrict__ W,
    const float* __restrict__ bias, const float* __restrict__ resid,
    float* __restrict__ Out, float alpha, int mode) {
  __shared__ __align__(16) float As[2 * ASZ];
  __shared__ __align__(16) float Bs[2 * BSZ];
  const int tid = threadIdx.x;
  const int wave = tid >> 5, lane = tid & 31;
  const int wr = wave >> 1, wc = wave & 1;
  const int lr = lane & 15, half = lane >> 4;
  const int n0 = blockIdx.x * 64;
  const int m0 = blockIdx.y * 64;
  v8f acc[2][2] = {};

  gemm_mainloop(X + (size_t)m0 * D_MODEL, D_MODEL, W + n0, D_MODEL,
                D_MODEL / 32, As, Bs, acc, tid, wr, wc, lr, half);

#pragma unroll
  for (int mi = 0; mi < 2; ++mi)
#pragma unroll
    for (int ni = 0; ni < 2; ++ni) {
      const int ncol = n0 + wc * 32 + ni * 16 + lr;
      const float bv = bias[ncol];
#pragma unroll
      for (int r = 0; r < 8; ++r) {
        const int mrow = m0 + wr * 32 + mi * 16 + r + half * 8;
        float v = acc[mi][ni][r] + bv;
        if (resid) v += resid[(size_t)mrow * D_MODEL + ncol];
        v *= alpha;
        const int b = mrow >> 11, s = mrow & (SEQ - 1);
        const int h = ncol >> 6, dk = ncol & 63;
        if (mode == 1) {
          Out[((size_t)(b * N_HEADS + h) * SEQ + s) * D_KH + dk] = v;
        } else if (mode == 2) {
          Out[((size_t)(b * N_HEADS + h) * D_KH + dk) * SEQ + s] = v;
        } else {
          Out[(size_t)mrow * D_MODEL + ncol] = v;
        }
      }
    }
}

// ---------------------------------------------------------------------------
// Raw scores per (b,h): Att[S,S] = Qs[S,64] @ Kt[64,S]  (Q pre-scaled, K is
// stored pre-transposed so the B tile is a contiguous async copy).
// ---------------------------------------------------------------------------
__global__ __launch_bounds__(128) void scores_kernel(
    const float* __restrict__ Q, const float* __restrict__ Kt,
    float* __restrict__ Att) {
  __shared__ __align__(16) float As[2 * ASZ];
  __shared__ __align__(16) float Bs[2 * BSZ];
  const int tid = threadIdx.x;
  const int wave = tid >> 5, lane = tid & 31;
  const int wr = wave >> 1, wc = wave & 1;
  const int lr = lane & 15, half = lane >> 4;
  const int n0 = blockIdx.x * 64;
  const int m0 = blockIdx.y * 64;
  const size_t bh = blockIdx.z;
  const float* Qb = Q + bh * (size_t)SEQ * D_KH;     // [S,64]
  const float* Kb = Kt + bh * (size_t)D_KH * SEQ;    // [64,S]
  v8f acc[2][2] = {};

  gemm_mainloop(Qb + (size_t)m0 * D_KH, D_KH, Kb + n0, SEQ, D_KH / 32, As, Bs,
                acc, tid, wr, wc, lr, half);

  float* Ab = Att + bh * (size_t)SEQ * SEQ;
#pragma unroll
  for (int mi = 0; mi < 2; ++mi)
#pragma unroll
    for (int ni = 0; ni < 2; ++ni) {
      const int ncol = n0 + wc * 32 + ni * 16 + lr;
#pragma unroll
      for (int r = 0; r < 8; ++r) {
        const int mrow = m0 + wr * 32 + mi * 16 + r + half * 8;
        Ab[(size_t)mrow * SEQ + ncol] = acc[mi][ni][r];
      }
    }
}

// ---------------------------------------------------------------------------
// In-place masked softmax over rows of Att (row length SEQ=2048).
// ---------------------------------------------------------------------------
__device__ __forceinline__ float wave_rsum(float v) {
#pragma unroll
  for (int o = 16; o; o >>= 1) v += __shfl_xor(v, o, 32);
  return v;
}
__device__ __forceinline__ float wave_rmax(float v) {
#pragma unroll
  for (int o = 16; o; o >>= 1) v = fmaxf(v, __shfl_xor(v, o, 32));
  return v;
}

__global__ __launch_bounds__(256) void softmax_mask_kernel(
    float* __restrict__ Att, const int* __restrict__ mask) {
  const size_t row = blockIdx.x;  // 0 .. B*H*S-1
  const int b = (int)(row / ((size_t)N_HEADS * SEQ));
  float* p = Att + row * (size_t)SEQ;
  const int* mrow = mask + (size_t)b * SEQ;
  const int tid = threadIdx.x;
  __shared__ float red[8];

  float vals[8];
  float lmax = -3.0e38f;
#pragma unroll
  for (int j = 0; j < 8; ++j) {
    const int c = tid + j * 256;
    float v = p[c];
    if (mrow[c] == 0) v = NEGVAL;
    vals[j] = v;
    lmax = fmaxf(lmax, v);
  }
  lmax = wave_rmax(lmax);
  if ((tid & 31) == 0) red[tid >> 5] = lmax;
  __syncthreads();
  float gmax = red[0];
#pragma unroll
  for (int i = 1; i < 8; ++i) gmax = fmaxf(gmax, red[i]);
  __syncthreads();

  float lsum = 0.f;
#pragma unroll
  for (int j = 0; j < 8; ++j) {
    vals[j] = expf(vals[j] - gmax);
    lsum += vals[j];
  }
  lsum = wave_rsum(lsum);
  if ((tid & 31) == 0) red[tid >> 5] = lsum;
  __syncthreads();
  float gsum = 0.f;
#pragma unroll
  for (int i = 0; i < 8; ++i) gsum += red[i];
  const float inv = 1.0f / gsum;
#pragma unroll
  for (int j = 0; j < 8; ++j) p[tid + j * 256] = vals[j] * inv;
}

// ---------------------------------------------------------------------------
// Context per (b,h): Att[S,S] @ V[S,64] -> Ctx in (B,S,D) layout.
// ---------------------------------------------------------------------------
__global__ __launch_bounds__(128) void av_gemm_kernel(
    const float* __restrict__ Att, const float* __restrict__ V,
    float* __restrict__ Ctx) {
  __shared__ __align__(16) float As[2 * ASZ];
  __shared__ __align__(16) float Bs[2 * BSZ];
  const int tid = threadIdx.x;
  const int wave = tid >> 5, lane = tid & 31;
  const int wr = wave >> 1, wc = wave & 1;
  const int lr = lane & 15, half = lane >> 4;
  const int m0 = blockIdx.y * 64;
  const size_t bh = blockIdx.z;
  const int b = (int)(bh >> 4), h = (int)(bh & 15);
  const float* Ab = Att + bh * (size_t)SEQ * SEQ;   // [S,S]
  const float* Vb = V + bh * (size_t)SEQ * D_KH;    // [S,64]
  v8f acc[2][2] = {};

  gemm_mainloop(Ab + (size_t)m0 * SEQ, SEQ, Vb, D_KH, SEQ / 32, As, Bs, acc,
                tid, wr, wc, lr, half);

#pragma unroll
  for (int mi = 0; mi < 2; ++mi)
#pragma unroll
    for (int ni = 0; ni < 2; ++ni) {
      const int ncol = wc * 32 + ni * 16 + lr;  // 0..63
#pragma unroll
      for (int r = 0; r < 8; ++r) {
        const int s = m0 + wr * 32 + mi * 16 + r + half * 8;
        Ctx[((size_t)b * SEQ + s) * D_MODEL + h * D_KH + ncol] = acc[mi][ni][r];
      }
    }
}

// ---------------------------------------------------------------------------
// Row-wise LayerNorm over D_MODEL=1024 with gamma/beta -> final out.
// ---------------------------------------------------------------------------
__global__ __launch_bounds__(256) void layernorm_kernel(
    const float* __restrict__ Xp, const float* __restrict__ gamma,
    const float* __restrict__ beta, float* __restrict__ Out) {
  const size_t row = blockIdx.x;  // 0 .. B*S-1
  const float* p = Xp + row * (size_t)D_MODEL;
  const int tid = threadIdx.x;
  __shared__ float reds[8], redq[8];

  float vals[4];
  float s = 0.f, sq = 0.f;
#pragma unroll
  for (int j = 0; j < 4; ++j) {
    const int c = tid + j * 256;
    const float v = p[c];
    vals[j] = v;
    s += v;
    sq += v * v;
  }
  s = wave_rsum(s);
  sq = wave_rsum(sq);
  if ((tid & 31) == 0) {
    reds[tid >> 5] = s;
    redq[tid >> 5] = sq;
  }
  __syncthreads();
  float gs = 0.f, gq = 0.f;
#pragma unroll
  for (int i = 0; i < 8; ++i) {
    gs += reds[i];
    gq += redq[i];
  }
  const float mu = gs * (1.0f / D_MODEL);
  const float var = gq * (1.0f / D_MODEL) - mu * mu;
  const float inv = rsqrtf(var + LN_EPS);
#pragma unroll
  for (int j = 0; j < 4; ++j) {
    const int c = tid + j * 256;
    Out[row * (size_t)D_MODEL + c] = (vals[j] - mu) * inv * gamma[c] + beta[c];
  }
}

// ---------------------------------------------------------------------------
extern "C" void kernel_launch(void* const* d_in, const int* in_sizes, int n_in,
                              void* d_out, int out_size, void* d_ws,
                              size_t ws_size, hipStream_t stream) {
  const float* x     = (const float*)d_in[0];
  const int*   mask  = (const int*)d_in[1];
  const float* Wq    = (const float*)d_in[2];
  const float* bq    = (const float*)d_in[3];
  const float* Wk    = (const float*)d_in[4];
  const float* bk    = (const float*)d_in[5];
  const float* Wv    = (const float*)d_in[6];
  const float* bv    = (const float*)d_in[7];
  const float* Wo    = (const float*)d_in[8];
  const float* bo    = (const float*)d_in[9];
  const float* gamma = (const float*)d_in[10];
  const float* beta  = (const float*)d_in[11];

  const size_t BSD = (size_t)BATCH * SEQ * D_MODEL;  // 8,388,608
  float* out = (float*)d_out;
  float* att = out + BSD;  // (B,H,S,S) region of d_out

  float* ws  = (float*)d_ws;
  float* Q   = ws;            // [B,H,S,64]
  float* Kt  = ws + BSD;      // [B,H,64,S] (pre-transposed)
  float* V   = ws + 2 * BSD;  // [B,H,S,64]
  float* Ctx = ws + 3 * BSD;  // [B,S,D]
  float* Pre = Q;             // reuse Q region for pre-LN output

  const dim3 blk(128);
  const dim3 gproj(D_MODEL / 64, (BATCH * SEQ) / 64);   // 16 x 128
  const dim3 gsc(SEQ / 64, SEQ / 64, BATCH * N_HEADS);  // 32 x 32 x 64
  const dim3 gav(1, SEQ / 64, BATCH * N_HEADS);         // 1 x 32 x 64

  const float inv_scale = 0.125f;  // 1/sqrt(D_KH)

  proj_gemm_kernel<<<gproj, blk, 0, stream>>>(x, Wq, bq, nullptr, Q, inv_scale, 1);
  proj_gemm_kernel<<<gproj, blk, 0, stream>>>(x, Wk, bk, nullptr, Kt, 1.0f, 2);
  proj_gemm_kernel<<<gproj, blk, 0, stream>>>(x, Wv, bv, nullptr, V, 1.0f, 1);

  scores_kernel<<<gsc, blk, 0, stream>>>(Q, Kt, att);

  softmax_mask_kernel<<<dim3(BATCH * N_HEADS * SEQ), dim3(256), 0, stream>>>(att, mask);

  av_gemm_kernel<<<gav, blk, 0, stream>>>(att, V, Ctx);

  proj_gemm_kernel<<<gproj, blk, 0, stream>>>(Ctx, Wo, bo, x, Pre, 1.0f, 0);

  layernorm_kernel<<<dim3(BATCH * SEQ), dim3(256), 0, stream>>>(Pre, gamma, beta, out);
}